// VQC_73813307949139
// MI455X (gfx1250) — compile-verified
//
#include <hip/hip_runtime.h>
#include <cstdint>

#define NCONCEPTS 1024

// ---------------------------------------------------------------------------
// Kernel 1: fold each concept (phi2, theta2, omega2) into the closed form's
// per-concept constants. omega2 is a global phase on the measured qubit and
// drops out of |<0|psi>|^2, so we only need:
//   { 0.5*cos(theta2), 0.5*sin(theta2), phi2, pad }
// 1024 concepts * 16 B = 16 KB table in workspace (L2-resident).
// ---------------------------------------------------------------------------
__global__ __launch_bounds__(256) void vqc_concept_pre(
    const float* __restrict__ w, float4* __restrict__ table) {
  int c = blockIdx.x * blockDim.x + threadIdx.x;
  if (c < NCONCEPTS) {
    float phi   = w[3 * c + 0];
    float theta = w[3 * c + 1];
    float s, cth;
    __sincosf(theta, &s, &cth);
    table[c] = make_float4(0.5f * cth, 0.5f * s, phi, 0.0f);
  }
}

// ---------------------------------------------------------------------------
// Kernel 2: main pass. Each thread handles 4 consecutive (b,d) elements:
//   3x global_load_b128 for 12 angles, 1x b128 for 4 indices,
//   LDS gather of float4 concept constants, 1x b128 store.
// The 16 KB concept LUT is staged into LDS with CDNA5 async global->LDS
// (ASYNCcnt path), then gathered with ds_load_b128.
//
// out = 0.5 + 0.5*cos(t1)*cos(t2) - 0.5*sin(t1)*sin(t2)*cos(w1 + p2)
//     = fmaf(cos(t1), lut.x, 0.5) - sin(t1)*lut.y*cos(w1 + lut.z)
// ---------------------------------------------------------------------------
__global__ __launch_bounds__(256) void vqc_main(
    const float* __restrict__ inst, const int* __restrict__ cidx,
    const float4* __restrict__ table, float* __restrict__ out,
    long long nquads) {
  __shared__ float4 lut[NCONCEPTS];  // 16 KB

  // --- async stage: 256 threads x 4 chunks x 16 B = 16 KB ---
  {
    unsigned lds_base = (unsigned)(uintptr_t)(&lut[0]);  // low 32b of LDS aperture = LDS offset
#pragma unroll
    for (int i = 0; i < 4; ++i) {
      unsigned byte_off = (threadIdx.x + i * 256u) * 16u;
      unsigned lds_addr = lds_base + byte_off;
      asm volatile("global_load_async_to_lds_b128 %0, %1, %2"
                   :: "v"(lds_addr), "v"(byte_off), "s"(table)
                   : "memory");
    }
  }
  asm volatile("s_wait_asynccnt 0x0" ::: "memory");
  __syncthreads();

  long long q = (long long)blockIdx.x * blockDim.x + threadIdx.x;
  if (q >= nquads) return;

  // 4 elements = 12 floats = 48 contiguous bytes
  const float4* a = reinterpret_cast<const float4*>(inst) + q * 3;
  float4 a0 = a[0];
  float4 a1 = a[1];
  float4 a2 = a[2];
  int4 id = reinterpret_cast<const int4*>(cidx)[q];

  // element layout: [p0 t0 w0 | p1 t1 w1 | p2 t2 w2 | p3 t3 w3]
  float th[4] = {a0.y, a1.x, a1.w, a2.z};
  float om[4] = {a0.z, a1.y, a2.x, a2.w};
  int   ic[4] = {id.x, id.y, id.z, id.w};
  float r[4];

#pragma unroll
  for (int e = 0; e < 4; ++e) {
    float4 wv = lut[ic[e] & (NCONCEPTS - 1)];  // ds_load_b128 gather
    float st, ct;
    __sincosf(th[e], &st, &ct);
    r[e] = fmaf(ct, wv.x, 0.5f) - st * wv.y * __cosf(om[e] + wv.z);
  }

  reinterpret_cast<float4*>(out)[q] = make_float4(r[0], r[1], r[2], r[3]);
}

// ---------------------------------------------------------------------------
// Launch: inputs are (instance f32 [B,D,3], concept_index i32 [B,D],
// concept_weights f32 [NC,3]); output f32 [B,D].
// ---------------------------------------------------------------------------
extern "C" void kernel_launch(void* const* d_in, const int* in_sizes, int n_in,
                              void* d_out, int out_size, void* d_ws, size_t ws_size,
                              hipStream_t stream) {
  const float* instance = (const float*)d_in[0];
  const int*   cidx     = (const int*)d_in[1];
  const float* cweights = (const float*)d_in[2];
  float4* table = (float4*)d_ws;  // 16 KB used

  vqc_concept_pre<<<NCONCEPTS / 256, 256, 0, stream>>>(cweights, table);

  long long n = (long long)out_size;   // B*D = 16,777,216
  long long nquads = n >> 2;           // 4 elements per thread
  int block = 256;
  long long grid = (nquads + block - 1) / block;
  vqc_main<<<(dim3)(unsigned)grid, block, 0, stream>>>(instance, cidx, table,
                                                       (float*)d_out, nquads);
}